// ZBL_75282186764710
// MI455X (gfx1250) — compile-verified
//
#include <hip/hip_runtime.h>
#include <hip/hip_bf16.h>

#define ZBL_KE 14.399645351950548f
#define MAX_ELEMS 128

typedef float v2f __attribute__((ext_vector_type(2)));
typedef float v8f __attribute__((ext_vector_type(8)));

// ---------------------------------------------------------------- zero fill
__global__ void zbl_zero(float* __restrict__ p, int n) {
    int i = blockIdx.x * blockDim.x + threadIdx.x;
    if (i < n) p[i] = 0.0f;
}

// ---------------------------------------------------------------- edge kernel
// Per edge: pair energy + analytic d(pair)/dr (matches the JAX VJP).
// - forces: global_atomic_add_f32 directly into d_out (100k-way scatter)
// - energy (nG bins) + raw stress sums (nG*9 bins): ds_add_f32 into LDS bins,
//   then either written as per-block partials to ws (primary, reduced by WMMA)
//   or atomically accumulated into d_out (fallback).
__global__ __launch_bounds__(256) void zbl_edge(
    const float* __restrict__ pos,    const float* __restrict__ shift,
    const float* __restrict__ a4,     const float* __restrict__ b4,
    const float* __restrict__ a0p,    const float* __restrict__ pp,
    const float* __restrict__ cr,     const float* __restrict__ cutp,
    const int*   __restrict__ num,    const int*   __restrict__ ei,
    const int*   __restrict__ batch,
    int nE, int nG, int nElems,
    float* __restrict__ f_out,        // d_out + nG  (forces, zeroed beforehand)
    float* __restrict__ P,            // primary: partials [gridDim.x][nG*10]; else null
    float* __restrict__ eacc,         // fallback: d_out (energy bins); else null
    float* __restrict__ sacc)         // fallback: d_out stress region; else null
{
    const int binsCount = nG * 10;
    __shared__ float bins[64 * 10];       // [0,nG): energy   [nG, nG*10): stress (g*9+k)
    __shared__ float s_cr[MAX_ELEMS];
    __shared__ float s_ab[8];

    const int tid = threadIdx.x;
    for (int j = tid; j < binsCount; j += blockDim.x) bins[j] = 0.0f;
    for (int j = tid; j < nElems && j < MAX_ELEMS; j += blockDim.x) s_cr[j] = cr[j];
    if (tid < 4) { s_ab[tid] = a4[tid]; s_ab[4 + tid] = b4[tid]; }
    __syncthreads();

    const float a0v  = a0p[0];
    const float pv   = pp[0];
    const float cutv = cutp[0];

    const int stride = gridDim.x * blockDim.x;
    for (int i = blockIdx.x * blockDim.x + tid; i < nE; i += stride) {
        // prefetch next grid-stride iteration's streamed edge data (gfx1250 global_prefetch)
        __builtin_prefetch(&ei[i + stride], 0, 1);
        __builtin_prefetch(&ei[nE + i + stride], 0, 1);
        __builtin_prefetch(&shift[3 * (i + stride)], 0, 1);

        const int s  = ei[i];
        const int d2 = ei[nE + i];
        const int zs = num[s];
        const int zd = num[d2];
        const float zi = (float)zs;
        const float zj = (float)zd;
        const float rc = fminf(cutv, s_cr[zs] + s_cr[zd]);

        const float vx = pos[3 * d2 + 0] - pos[3 * s + 0] + shift[3 * i + 0];
        const float vy = pos[3 * d2 + 1] - pos[3 * s + 1] + shift[3 * i + 1];
        const float vz = pos[3 * d2 + 2] - pos[3 * s + 2] + shift[3 * i + 2];
        const float r  = sqrtf(vx * vx + vy * vy + vz * vz);

        const float aij = a0v / (powf(zi, pv) + powf(zj, pv));
        const float x   = r / aij;

        float phi = 0.0f, dphidx = 0.0f;
#pragma unroll
        for (int k = 0; k < 4; ++k) {
            const float bk = s_ab[4 + k];
            const float t  = s_ab[k] * expf(bk * x);
            phi    += t;
            dphidx += bk * t;
        }

        const float ec  = ZBL_KE * zi * zj / r;     // screened Coulomb
        const float dec = -ec / r;                  // d(ec)/dr

        const float xr = r / rc;
        float env = 0.0f, denv = 0.0f;              // envelope (p=6) and d/dxr
        if (xr < 1.0f) {
            const float xr2 = xr * xr;
            const float xr4 = xr2 * xr2;
            const float xr5 = xr4 * xr;
            const float xr6 = xr5 * xr;
            const float xr7 = xr6 * xr;
            const float xr8 = xr7 * xr;
            env  = 1.0f - 28.0f * xr6 + 48.0f * xr7 - 21.0f * xr8;
            denv = -168.0f * xr5 + 336.0f * xr6 - 168.0f * xr7;
        }

        const float pair  = ec * phi * env;
        const float dpair = dec * phi * env
                          + ec * (dphidx / aij) * env
                          + ec * phi * (denv / rc);
        const float gs = 0.5f * dpair / r;          // egradij = gs * v
        const float gx = gs * vx, gy = gs * vy, gz = gs * vz;

        // energies[batch[dst]] += 0.5*pair
        unsafeAtomicAdd(&bins[batch[d2]], 0.5f * pair);

        // forces: +g at src, -g at dst   (forces = -egradi)
        unsafeAtomicAdd(&f_out[3 * s  + 0],  gx);
        unsafeAtomicAdd(&f_out[3 * s  + 1],  gy);
        unsafeAtomicAdd(&f_out[3 * s  + 2],  gz);
        unsafeAtomicAdd(&f_out[3 * d2 + 0], -gx);
        unsafeAtomicAdd(&f_out[3 * d2 + 1], -gy);
        unsafeAtomicAdd(&f_out[3 * d2 + 2], -gz);

        // raw stress sums per batch[src]: sum v_x * g_y  (scaled by 0.5/vol later)
        float* sb = &bins[nG + batch[s] * 9];
        unsafeAtomicAdd(&sb[0], vx * gx); unsafeAtomicAdd(&sb[1], vx * gy); unsafeAtomicAdd(&sb[2], vx * gz);
        unsafeAtomicAdd(&sb[3], vy * gx); unsafeAtomicAdd(&sb[4], vy * gy); unsafeAtomicAdd(&sb[5], vy * gz);
        unsafeAtomicAdd(&sb[6], vz * gx); unsafeAtomicAdd(&sb[7], vz * gy); unsafeAtomicAdd(&sb[8], vz * gz);
    }
    __syncthreads();

    if (P) {
        float* row = P + (size_t)blockIdx.x * binsCount;
        for (int j = tid; j < binsCount; j += blockDim.x) row[j] = bins[j];
    } else {
        for (int j = tid; j < binsCount; j += blockDim.x) {
            const float v = bins[j];
            if (j < nG) unsafeAtomicAdd(&eacc[j], v);
            else        unsafeAtomicAdd(&sacc[j - nG], v);
        }
    }
}

// ---------------------------------------------------------------- WMMA reduce
// out_col[c] = sum over NB partial rows of P[row][c], computed as D = 1(16x4) x B + D
// with V_WMMA_F32_16X16X4_F32. A = all-ones => result is the exact f32 column
// sums and is invariant to K-row permutation inside the B fragment.
// Out-of-range lanes load a clamped (valid) address and mask by 0; any garbage
// stays confined to that lane's D column, which is never stored.
// Finalize: energies written raw; stress scaled by 0.5/det(cell).
__global__ __launch_bounds__(32) void zbl_reduce_wmma(
    const float* __restrict__ P, int NB, int binsCount,
    const float* __restrict__ cell, int nG,
    float* __restrict__ e_out,        // d_out
    float* __restrict__ s_out)        // d_out + nG + 3*nN
{
    const int lane  = threadIdx.x;            // wave32: one wave per block
    const int col0  = blockIdx.x * 16;
    const int half  = lane >> 4;              // 0: lanes 0-15, 1: lanes 16-31
    const int n     = col0 + (lane & 15);
    const bool nok  = (n < binsCount);
    const float msk = nok ? 1.0f : 0.0f;

    // clamped column pointer; non-divergent strided loads from here on
    const float* colp = P + (nok ? n : 0);
    const int off0 = (half ? 2 : 0) * binsCount;   // rows k0+{0,2}
    const int off1 = (half ? 3 : 1) * binsCount;   // rows k0+{1,3}
    const int rowStep = 4 * binsCount;             // elements per K-chunk

    v8f d = {0.f, 0.f, 0.f, 0.f, 0.f, 0.f, 0.f, 0.f};
    const v2f a = {1.0f, 1.0f};               // A matrix: 16x4 of ones

#pragma unroll 4
    for (int k0 = 0; k0 < NB; k0 += 4) {
        v2f b;
        b.x = msk * colp[off0];
        b.y = msk * colp[off1];
        colp += rowStep;
        d = __builtin_amdgcn_wmma_f32_16x16x4_f32(
                /*neg_a=*/false, a, /*neg_b=*/false, b,
                /*c_mod=*/(short)0, d, /*reuse_a=*/false, /*reuse_b=*/false);
    }

    // D row M=0 lives in VGPR 0 on lanes 0-15 (N = lane); all rows are equal.
    if (lane < 16) {
        const int c = col0 + lane;
        if (c < binsCount) {
            const float sum = d[0];
            if (c < nG) {
                e_out[c] = sum;
            } else {
                const int sc = c - nG;
                const int g  = sc / 9;
                const float* C = cell + g * 9;
                const float det = C[0] * (C[4] * C[8] - C[5] * C[7])
                                - C[1] * (C[3] * C[8] - C[5] * C[6])
                                + C[2] * (C[3] * C[7] - C[4] * C[6]);
                s_out[sc] = 0.5f * sum / det;
            }
        }
    }
}

// ---------------------------------------------------------------- fallback stress scale
__global__ void zbl_scale_stress(float* __restrict__ s_out,
                                 const float* __restrict__ cell, int nG) {
    int i = blockIdx.x * blockDim.x + threadIdx.x;
    if (i < nG * 9) {
        const int g = i / 9;
        const float* C = cell + g * 9;
        const float det = C[0] * (C[4] * C[8] - C[5] * C[7])
                        - C[1] * (C[3] * C[8] - C[5] * C[6])
                        + C[2] * (C[3] * C[7] - C[4] * C[6]);
        s_out[i] = 0.5f * s_out[i] / det;
    }
}

// ---------------------------------------------------------------- launch
extern "C" void kernel_launch(void* const* d_in, const int* in_sizes, int n_in,
                              void* d_out, int out_size, void* d_ws, size_t ws_size,
                              hipStream_t stream) {
    const float* pos   = (const float*)d_in[0];
    const float* shift = (const float*)d_in[1];
    const float* a4    = (const float*)d_in[2];
    const float* b4    = (const float*)d_in[3];
    const float* a0    = (const float*)d_in[4];
    const float* p     = (const float*)d_in[5];
    const float* cr    = (const float*)d_in[6];
    const float* cell  = (const float*)d_in[7];
    const float* cut   = (const float*)d_in[8];
    const int*   num   = (const int*)d_in[9];
    const int*   ei    = (const int*)d_in[10];
    const int*   batch = (const int*)d_in[11];

    const int nN     = in_sizes[0] / 3;
    const int nE     = in_sizes[10] / 2;
    const int nG     = in_sizes[7] / 9;
    const int nElems = in_sizes[6];
    const int bins   = nG * 10;

    float* out   = (float*)d_out;
    float* f_out = out + nG;
    float* s_out = out + nG + 3 * nN;

    // Deterministic path choice from the fixed ws_size.
    const size_t rowBytes = (size_t)bins * sizeof(float);
    int NB = (rowBytes > 0) ? (int)((ws_size / rowBytes < 1024) ? (ws_size / rowBytes) : 1024) : 0;
    NB &= ~3;                       // multiple of 4 for K-chunking
    const bool primary = (NB >= 8);

    if (primary) {
        // zero only forces; energies/stress fully written by the WMMA reduce
        int nF = 3 * nN;
        zbl_zero<<<(nF + 255) / 256, 256, 0, stream>>>(f_out, nF);
        zbl_edge<<<NB, 256, 0, stream>>>(pos, shift, a4, b4, a0, p, cr, cut,
                                         num, ei, batch, nE, nG, nElems,
                                         f_out, (float*)d_ws, nullptr, nullptr);
        zbl_reduce_wmma<<<(bins + 15) / 16, 32, 0, stream>>>(
            (const float*)d_ws, NB, bins, cell, nG, out, s_out);
    } else {
        zbl_zero<<<(out_size + 255) / 256, 256, 0, stream>>>(out, out_size);
        zbl_edge<<<1024, 256, 0, stream>>>(pos, shift, a4, b4, a0, p, cr, cut,
                                           num, ei, batch, nE, nG, nElems,
                                           f_out, nullptr, out, s_out);
        zbl_scale_stress<<<(nG * 9 + 255) / 256, 256, 0, stream>>>(s_out, cell, nG);
    }
}